// LPRON_69870527971647
// MI455X (gfx1250) — compile-verified
//
#include <hip/hip_runtime.h>
#include <hip/hip_bf16.h>

typedef __attribute__((ext_vector_type(2))) float v2f;
typedef __attribute__((ext_vector_type(8))) float v8f;

constexpr int S_ = 4;
constexpr int F_ = 1128;
constexpr int T_ = 1000;
constexpr int D_ = 13;
constexpr int H_ = 1000;
constexpr float EPS_COV = 1e-3f;
constexpr float EPS_KL  = 1e-5f;
constexpr float BN_EPS  = 1e-5f;

// ---------------------------------------------------------------------------
// Gauss-Jordan inversion of a 13x13 SPD matrix living in LDS (lane-0 serial).
// ---------------------------------------------------------------------------
__device__ __forceinline__ void invert13(const float (*src)[16], float (*inv)[16],
                                         float (*work)[16]) {
    for (int i = 0; i < D_; ++i)
        for (int j = 0; j < D_; ++j) {
            work[i][j] = src[i][j];
            inv[i][j] = (i == j) ? 1.f : 0.f;
        }
    for (int k = 0; k < D_; ++k) {
        float piv = 1.f / work[k][k];
        for (int j = 0; j < D_; ++j) { work[k][j] *= piv; inv[k][j] *= piv; }
        for (int i = 0; i < D_; ++i) {
            if (i == k) continue;
            float f = work[i][k];
            for (int j = 0; j < D_; ++j) {
                work[i][j] -= f * work[k][j];
                inv[i][j]  -= f * inv[k][j];
            }
        }
    }
}

// ---------------------------------------------------------------------------
// Kernel 1: per (s,f) pair, compute Gram/mean via V_WMMA_F32_16X16X4_F32,
// then covariances, symmetric KL, and the masked log feature.
// One wave (32 lanes) per pair; grid = S*F.
// A-matrix 16x4 f32 layout: lane m (0-15) vgpr0=K0, vgpr1=K1; lanes 16-31
// carry K2/K3.  For a Gram matrix the B operand has the identical per-lane
// contents, so a single float2 load feeds both A and B.
// ---------------------------------------------------------------------------
__global__ __launch_bounds__(32) void moments_kl_kernel(
    const float* __restrict__ P, const float* __restrict__ Q,
    const float* __restrict__ plen, const float* __restrict__ qlen,
    const float* __restrict__ mask, float* __restrict__ feats)
{
    const int pair = blockIdx.x;                 // s*F + f
    const float* Xp = P + (size_t)pair * (T_ * D_);
    const float* Xq = Q + (size_t)pair * (T_ * D_);
    const int lane = threadIdx.x;
    const int m  = lane & 15;
    const int hi = lane >> 4;
    const int mm = (m < D_) ? m : 0;             // clamp for in-bounds load
    const bool mv = (m < D_);

    v8f accp = {0.f,0.f,0.f,0.f,0.f,0.f,0.f,0.f};
    v8f accq = {0.f,0.f,0.f,0.f,0.f,0.f,0.f,0.f};
    float csp = 0.f, csq = 0.f;                  // column sums -> means

    for (int t0 = 0; t0 < T_; t0 += 4) {         // T_ % 4 == 0, no tail
        const int r0 = t0 + 2 * hi;
        float ax = Xp[r0 * D_ + mm];
        float ay = Xp[(r0 + 1) * D_ + mm];
        ax = mv ? ax : 0.f;  ay = mv ? ay : 0.f;
        v2f a;  a.x = ax;  a.y = ay;
        accp = __builtin_amdgcn_wmma_f32_16x16x4_f32(
            false, a, false, a, (short)0, accp, false, false);
        csp += ax + ay;

        float bx = Xq[r0 * D_ + mm];
        float by = Xq[(r0 + 1) * D_ + mm];
        bx = mv ? bx : 0.f;  by = mv ? by : 0.f;
        v2f b;  b.x = bx;  b.y = by;
        accq = __builtin_amdgcn_wmma_f32_16x16x4_f32(
            false, b, false, b, (short)0, accq, false, false);
        csq += bx + by;
    }

    __shared__ float gp[16][16], gq[16][16];     // Gram -> covariance (in place)
    __shared__ float ip[16][16], iq[16][16];     // inverses
    __shared__ float sc[16][16];                 // Gauss-Jordan scratch
    __shared__ float cs[64];                     // column-sum halves p|q

    // C/D layout: vgpr r, lanes 0-15 -> M=r, lanes 16-31 -> M=r+8; N = lane%16
    const int row0 = hi * 8;
    #pragma unroll
    for (int r = 0; r < 8; ++r) {
        gp[row0 + r][m] = accp[r];
        gq[row0 + r][m] = accq[r];
    }
    cs[lane]      = csp;
    cs[32 + lane] = csq;
    __syncthreads();

    if (lane == 0) {
        const float lp = plen[pair];
        const float lq = qlen[pair];
        float meanp[D_], meanq[D_];
        for (int i = 0; i < D_; ++i) {
            meanp[i] = (cs[i]      + cs[16 + i]) / lp;
            meanq[i] = (cs[32 + i] + cs[48 + i]) / lq;
        }
        for (int i = 0; i < D_; ++i)
            for (int j = 0; j < D_; ++j) {
                float e = (i == j) ? EPS_COV : 0.f;
                gp[i][j] = gp[i][j] / lp - meanp[i] * meanp[j] + e;
                gq[i][j] = gq[i][j] / lq - meanq[i] * meanq[j] + e;
            }
        invert13(gq, iq, sc);
        invert13(gp, ip, sc);

        float tr = 0.f, quad = 0.f;
        for (int i = 0; i < D_; ++i) {
            float di = meanp[i] - meanq[i];
            float ui = 0.f;
            for (int j = 0; j < D_; ++j) {
                tr += iq[i][j] * gp[j][i] + ip[i][j] * gq[j][i];
                ui += (iq[i][j] + ip[i][j]) * (meanp[j] - meanq[j]);
            }
            quad += di * ui;
        }
        float kl   = 0.25f * (tr + quad - 2.f * (float)D_);
        float feat = logf(kl + EPS_KL);
        feat = (feat + 1.f) * mask[pair] - 1.f;
        feats[pair] = feat;
    }
}

// ---------------------------------------------------------------------------
// Kernel 2: BatchNorm1d over the batch axis (S=4), biased variance.
// ---------------------------------------------------------------------------
__global__ void bn_kernel(const float* __restrict__ feats,
                          const float* __restrict__ gamma,
                          const float* __restrict__ beta,
                          float* __restrict__ xout)
{
    int f = blockIdx.x * blockDim.x + threadIdx.x;
    if (f >= F_) return;
    float v[S_];
    float mu = 0.f;
    #pragma unroll
    for (int s = 0; s < S_; ++s) { v[s] = feats[s * F_ + f]; mu += v[s]; }
    mu *= (1.f / (float)S_);
    float var = 0.f;
    #pragma unroll
    for (int s = 0; s < S_; ++s) { float d = v[s] - mu; var += d * d; }
    var *= (1.f / (float)S_);
    float inv = 1.f / sqrtf(var + BN_EPS);
    float g = gamma[f], b = beta[f];
    #pragma unroll
    for (int s = 0; s < S_; ++s)
        xout[s * F_ + f] = (v[s] - mu) * inv * g + b;
}

// ---------------------------------------------------------------------------
// Kernel 3: one MLP layer  out[4,N] = act(A[4,K] @ W[K,N] + bias).
// One wave per 16-wide N tile; K streamed 4 at a time through f32 WMMA.
// K is always a multiple of 4 here (1128 / 1000).
// ---------------------------------------------------------------------------
__global__ __launch_bounds__(32) void mlp_layer_kernel(
    const float* __restrict__ A, const float* __restrict__ W,
    const float* __restrict__ bias, float* __restrict__ out,
    int K, int N, int do_relu)
{
    const int lane = threadIdx.x;
    const int m  = lane & 15;
    const int hi = lane >> 4;
    const int n0 = blockIdx.x * 16;
    const int col  = n0 + m;
    const int colc = (col < N) ? col : (N - 1);
    const bool cok = (col < N);
    const int  ms  = (m < S_) ? m : 0;
    const bool mok = (m < S_);

    v8f acc = {0.f,0.f,0.f,0.f,0.f,0.f,0.f,0.f};
    for (int k0 = 0; k0 < K; k0 += 4) {
        const int kx = k0 + 2 * hi;
        float ax = A[ms * K + kx];
        float ay = A[ms * K + kx + 1];
        ax = mok ? ax : 0.f;  ay = mok ? ay : 0.f;
        v2f a;  a.x = ax;  a.y = ay;

        float bx = W[(size_t)kx * N + colc];
        float by = W[(size_t)(kx + 1) * N + colc];
        bx = cok ? bx : 0.f;  by = cok ? by : 0.f;
        v2f b;  b.x = bx;  b.y = by;

        acc = __builtin_amdgcn_wmma_f32_16x16x4_f32(
            false, a, false, b, (short)0, acc, false, false);
    }

    if (hi == 0 && cok) {
        float bs = bias[colc];
        #pragma unroll
        for (int r = 0; r < S_; ++r) {          // rows 0..3 live in vgprs 0..3
            float vv = acc[r] + bs;
            if (do_relu) vv = fmaxf(vv, 0.f);
            out[r * N + col] = vv;
        }
    }
}

// ---------------------------------------------------------------------------
extern "C" void kernel_launch(void* const* d_in, const int* in_sizes, int n_in,
                              void* d_out, int out_size, void* d_ws, size_t ws_size,
                              hipStream_t stream)
{
    const float* p_vects   = (const float*)d_in[0];
    const float* q_vects   = (const float*)d_in[1];
    const float* p_lengths = (const float*)d_in[2];
    const float* q_lengths = (const float*)d_in[3];
    const float* mask      = (const float*)d_in[4];
    const float* gamma     = (const float*)d_in[5];
    const float* beta      = (const float*)d_in[6];
    const float* W1 = (const float*)d_in[7];  const float* b1 = (const float*)d_in[8];
    const float* W2 = (const float*)d_in[9];  const float* b2 = (const float*)d_in[10];
    const float* W3 = (const float*)d_in[11]; const float* b3 = (const float*)d_in[12];
    const float* W4 = (const float*)d_in[13]; const float* b4 = (const float*)d_in[14];
    const float* W5 = (const float*)d_in[15]; const float* b5 = (const float*)d_in[16];
    const float* W6 = (const float*)d_in[17]; const float* b6 = (const float*)d_in[18];
    const float* W7 = (const float*)d_in[19]; const float* b7 = (const float*)d_in[20];

    float* ws    = (float*)d_ws;
    float* feats = ws;                         // S*F
    float* xbn   = ws + S_ * F_;               // S*F
    float* hA    = ws + 2 * S_ * F_;           // S*H
    float* hB    = hA + S_ * H_;               // S*H

    // Phase 1: moments + symmetric KL + masked log feature (bandwidth bound:
    // ~470 MB of fp32 input -> ~20 us floor at 23.3 TB/s; Gram via f32 WMMA).
    moments_kl_kernel<<<dim3(S_ * F_), dim3(32), 0, stream>>>(
        p_vects, q_vects, p_lengths, q_lengths, mask, feats);

    // Phase 2: BatchNorm over the batch axis.
    bn_kernel<<<dim3((F_ + 127) / 128), dim3(128), 0, stream>>>(
        feats, gamma, beta, xbn);

    // Phase 3: MLP (weights ~25 MB, read once -> ~1.1 us bandwidth bound).
    const int nt = (H_ + 15) / 16;   // 63 tiles
    mlp_layer_kernel<<<dim3(nt), dim3(32), 0, stream>>>(xbn, W1, b1, hA, F_, H_, 1);
    mlp_layer_kernel<<<dim3(nt), dim3(32), 0, stream>>>(hA,  W2, b2, hB, H_, H_, 1);
    mlp_layer_kernel<<<dim3(nt), dim3(32), 0, stream>>>(hB,  W3, b3, hA, H_, H_, 1);
    mlp_layer_kernel<<<dim3(nt), dim3(32), 0, stream>>>(hA,  W4, b4, hB, H_, H_, 1);
    mlp_layer_kernel<<<dim3(nt), dim3(32), 0, stream>>>(hB,  W5, b5, hA, H_, H_, 1);
    mlp_layer_kernel<<<dim3(nt), dim3(32), 0, stream>>>(hA,  W6, b6, hB, H_, H_, 1);
    mlp_layer_kernel<<<dim3(1),  dim3(32), 0, stream>>>(hB,  W7, b7, (float*)d_out, H_, 1, 0);
}